// PostProcess_21148418965810
// MI455X (gfx1250) — compile-verified
//
#include <hip/hip_runtime.h>
#include <stdint.h>

// Problem constants (match reference)
#define NB   256          // batch
#define NQ   900          // queries
#define NC   256          // classes
#define KTOP 100
#define QC   (NQ * NC)    // 230400 per image

// Select machinery
#define NBINS   2048      // top 11 bits of positive float -> monotonic bins
#define NHSUB   4         // replicated sub-histograms (LDS atomic conflict relief)
#define CANDMAX 2048      // candidate cap per image

// Streaming pass geometry: 45 blocks/image * 20 rows/block, 4 rows (1024 el) per iter
#define ROWS_PER_BLK 20
#define NXBLK        (NQ / ROWS_PER_BLK)   // 45
#define ITERS        (ROWS_PER_BLK / 4)    // 5

typedef int   v4i __attribute__((ext_vector_type(4)));
typedef float v4f __attribute__((ext_vector_type(4)));
#define AS_GLOBAL __attribute__((address_space(1)))
#define AS_LDS    __attribute__((address_space(3)))

#if __has_builtin(__builtin_amdgcn_global_load_async_to_lds_b128)
#define HAVE_ASYNC 1
#else
#define HAVE_ASYNC 0
#endif

#if __has_builtin(__builtin_amdgcn_s_wait_asynccnt)
#define WAIT_ASYNC(n) __builtin_amdgcn_s_wait_asynccnt(n)
#else
#define WAIT_ASYNC(n) asm volatile("s_wait_asynccnt %0" ::"i"(n) : "memory")
#endif

__device__ __forceinline__ float fsig(float x) { return 1.0f / (1.0f + __expf(-x)); }

__device__ __forceinline__ unsigned binOf(float s) {
    unsigned b = __float_as_uint(s) >> 20;   // scores are >= 0, so this is monotone
    return b < NBINS ? b : (NBINS - 1);
}

// ---------------------------------------------------------------- pass 0: zero
__global__ void zero_kernel(unsigned* __restrict__ p, int n) {
    int i = blockIdx.x * blockDim.x + threadIdx.x;
    if (i < n) p[i] = 0u;
}

// ------------------------------------------------- pass 1: fused-score histogram
// grid (NXBLK, NB), block 256.  Logits staged global->LDS asynchronously
// (double buffered) via gfx1250 async-LDS instructions.  Histogram kept as
// 4 interleaved copies: hbin[bin*4 + (lane&3)] -> same-bin traffic from
// adjacent lanes hits different addresses AND different banks.
__global__ void score_hist_kernel(const float* __restrict__ logits,
                                  const float* __restrict__ obj,
                                  unsigned* __restrict__ hist) {
    __shared__ unsigned hbin[NBINS * NHSUB];   // 32 KB
#if HAVE_ASYNC
    __shared__ float stage[2][1024];           // 8 KB
#endif
    const int b    = blockIdx.y;
    const int t    = threadIdx.x;
    const int sub  = t & (NHSUB - 1);
    const int row0 = blockIdx.x * ROWS_PER_BLK;

    for (int i = t; i < NBINS * NHSUB; i += 256) hbin[i] = 0u;

    const float* base = logits + (size_t)b * QC + (size_t)row0 * NC;
    const float* objb = obj + (size_t)b * NQ;

#if HAVE_ASYNC
    __builtin_amdgcn_global_load_async_to_lds_b128(
        (AS_GLOBAL v4i*)(base + t * 4),
        (AS_LDS v4i*)(&stage[0][t * 4]), 0, 0);
#endif
    __syncthreads();   // hbin init visible (+ prologue issued)

    for (int it = 0; it < ITERS; ++it) {
        v4f v;
#if HAVE_ASYNC
        if (it + 1 < ITERS) {
            __builtin_amdgcn_global_load_async_to_lds_b128(
                (AS_GLOBAL v4i*)(base + (it + 1) * 1024 + t * 4),
                (AS_LDS v4i*)(&stage[(it + 1) & 1][t * 4]), 0, 0);
            WAIT_ASYNC(1);   // oldest (current buffer) complete; prefetch in flight
        } else {
            WAIT_ASYNC(0);
        }
        __syncthreads();     // all waves' portions of current buffer visible
        v = *reinterpret_cast<const v4f*>(&stage[it & 1][t * 4]);
#else
        v = *reinterpret_cast<const v4f*>(base + it * 1024 + t * 4);
#endif
        const int   q  = row0 + it * 4 + (t >> 6);
        const int   c0 = (t & 63) * 4;
        const float op = __expf(-objb[q]);

        float s0 = (c0 + 0 >= 240) ? 0.0f : op * fsig(v.x);
        float s1 = (c0 + 1 >= 240) ? 0.0f : op * fsig(v.y);
        float s2 = (c0 + 2 >= 240) ? 0.0f : op * fsig(v.z);
        float s3 = (c0 + 3 >= 240) ? 0.0f : op * fsig(v.w);

        atomicAdd(&hbin[binOf(s0) * NHSUB + sub], 1u);
        atomicAdd(&hbin[binOf(s1) * NHSUB + sub], 1u);
        atomicAdd(&hbin[binOf(s2) * NHSUB + sub], 1u);
        atomicAdd(&hbin[binOf(s3) * NHSUB + sub], 1u);

        __syncthreads();     // reads of stage done before its next refill
    }

    unsigned* gh = hist + (size_t)b * NBINS;
    for (int i = t; i < NBINS; i += 256) {
        unsigned cv = hbin[i * NHSUB + 0] + hbin[i * NHSUB + 1] +
                      hbin[i * NHSUB + 2] + hbin[i * NHSUB + 3];
        if (cv) atomicAdd(&gh[i], cv);
    }
}

// -------------------------------------------------- pass 2: threshold bin/image
__global__ void thresh_kernel(const unsigned* __restrict__ hist,
                              unsigned* __restrict__ thresh) {
    const int b = blockIdx.x;
    if (threadIdx.x == 0) {
        const unsigned* h = hist + (size_t)b * NBINS;
        unsigned cum = 0, th = 0;
        for (int bin = NBINS - 1; bin >= 0; --bin) {
            cum += h[bin];
            if (cum >= KTOP) { th = (unsigned)bin; break; }
        }
        thresh[b] = th;
    }
}

// ------------------------------------------------------ pass 3: compact >= thr
__global__ void compact_kernel(const float* __restrict__ logits,
                               const float* __restrict__ obj,
                               const unsigned* __restrict__ thresh,
                               unsigned* __restrict__ cnt,
                               float* __restrict__ cscore,
                               int* __restrict__ cidx) {
    const int      b    = blockIdx.y;
    const int      t    = threadIdx.x;
    const int      row0 = blockIdx.x * ROWS_PER_BLK;
    const unsigned th   = thresh[b];

    const float* base = logits + (size_t)b * QC + (size_t)row0 * NC;
    const float* objb = obj + (size_t)b * NQ;

    for (int it = 0; it < ITERS; ++it) {
        if (it + 1 < ITERS)
            __builtin_prefetch(base + (it + 1) * 1024 + t * 4, 0, 0);  // global_prefetch_b8
        // streaming read, no reuse: keep it out of the way of L2 (TH_NT)
        v4f v = __builtin_nontemporal_load(
            reinterpret_cast<const v4f*>(base + it * 1024 + t * 4));

        const int   q  = row0 + it * 4 + (t >> 6);
        const int   c0 = (t & 63) * 4;
        const float op = __expf(-objb[q]);

        float s[4];
        s[0] = (c0 + 0 >= 240) ? 0.0f : op * fsig(v.x);
        s[1] = (c0 + 1 >= 240) ? 0.0f : op * fsig(v.y);
        s[2] = (c0 + 2 >= 240) ? 0.0f : op * fsig(v.z);
        s[3] = (c0 + 3 >= 240) ? 0.0f : op * fsig(v.w);

#pragma unroll
        for (int j = 0; j < 4; ++j) {
            if (binOf(s[j]) >= th) {
                unsigned slot = atomicAdd(&cnt[b], 1u);
                if (slot < CANDMAX) {
                    cscore[(size_t)b * CANDMAX + slot] = s[j];
                    cidx[(size_t)b * CANDMAX + slot]   = q * NC + c0 + j;
                }
            }
        }
    }
}

// ------------------------------------- pass 4: bitonic top-K + box gather/scale
// grid NB, block 1024 (32 waves).  LDS: 16 KB of keys+ids.
__global__ void select_kernel(const float* __restrict__ boxes,
                              const float* __restrict__ tsizes,
                              const unsigned* __restrict__ cnt,
                              const float* __restrict__ cscore,
                              const int* __restrict__ cidx,
                              float* __restrict__ out) {
    __shared__ float ss[CANDMAX];
    __shared__ int   si[CANDMAX];
    const int b = blockIdx.x;
    const int t = threadIdx.x;

    unsigned n = cnt[b];
    if (n > CANDMAX) n = CANDMAX;

    for (int i = t; i < CANDMAX; i += 1024) {
        if ((unsigned)i < n) {
            ss[i] = cscore[(size_t)b * CANDMAX + i];
            si[i] = cidx[(size_t)b * CANDMAX + i];
        } else {
            ss[i] = -1.0f;
            si[i] = 0x7FFFFFFF;
        }
    }
    __syncthreads();

    // Bitonic sort, final order: descending score, ties by ascending flat index
    for (int k = 2; k <= CANDMAX; k <<= 1) {
        for (int j = k >> 1; j > 0; j >>= 1) {
            for (int i = t; i < CANDMAX; i += 1024) {
                int ixj = i ^ j;
                if (ixj > i) {
                    float a = ss[i], c = ss[ixj];
                    int   ai = si[i], ci = si[ixj];
                    bool  desc    = ((i & k) == 0);
                    bool  aLess   = (a < c) || (a == c && ai > ci);
                    bool  aGreat  = (a > c) || (a == c && ai < ci);
                    if (desc ? aLess : aGreat) {
                        ss[i] = c;  ss[ixj] = a;
                        si[i] = ci; si[ixj] = ai;
                    }
                }
            }
            __syncthreads();
        }
    }

    if (t < KTOP) {
        float sc  = ss[t];
        int   idx = si[t];
        int   q   = idx / NC;
        int   lab = idx - q * NC;
        if (q < 0 || q >= NQ) { q = 0; }          // pad safety (shouldn't occur)

        const float* bx = boxes + ((size_t)b * NQ + q) * 4;
        float cx = bx[0], cy = bx[1], w = bx[2], h = bx[3];
        float img_h = tsizes[b * 2 + 0];
        float img_w = tsizes[b * 2 + 1];

        float x0 = (cx - 0.5f * w) * img_w;
        float y0 = (cy - 0.5f * h) * img_h;
        float x1 = (cx + 0.5f * w) * img_w;
        float y1 = (cy + 0.5f * h) * img_h;

        // d_out: scores [B*K] | labels [B*K] | boxes [B*K*4], all f32
        float* outScores = out;
        float* outLabels = out + NB * KTOP;
        float* outBoxes  = out + 2 * NB * KTOP;

        outScores[b * KTOP + t] = sc;
        outLabels[b * KTOP + t] = (float)lab;
        float* ob = outBoxes + ((size_t)b * KTOP + t) * 4;
        ob[0] = x0; ob[1] = y0; ob[2] = x1; ob[3] = y1;
    }
}

// ------------------------------------------------------------------- launcher
extern "C" void kernel_launch(void* const* d_in, const int* in_sizes, int n_in,
                              void* d_out, int out_size, void* d_ws, size_t ws_size,
                              hipStream_t stream) {
    const float* logits = (const float*)d_in[0];   // (B,Q,C)
    const float* obj    = (const float*)d_in[1];   // (B,Q)
    const float* boxes  = (const float*)d_in[2];   // (B,Q,4)
    const float* tsz    = (const float*)d_in[3];   // (B,2)

    uint8_t* ws = (uint8_t*)d_ws;
    const size_t HIST_BYTES = (size_t)NB * NBINS * sizeof(unsigned);        // 2 MB
    unsigned* hist   = (unsigned*)(ws);
    unsigned* thresh = (unsigned*)(ws + HIST_BYTES);                        // 1 KB
    unsigned* cnt    = (unsigned*)(ws + HIST_BYTES + 1024);                 // 1 KB
    float*    cscore = (float*)   (ws + HIST_BYTES + 2048);                 // 2 MB
    int*      cidx   = (int*)     (ws + HIST_BYTES + 2048 +
                                   (size_t)NB * CANDMAX * sizeof(float));   // 2 MB

    const int nzero = (int)((HIST_BYTES + 2048) / sizeof(unsigned));        // 524800
    zero_kernel<<<(nzero + 255) / 256, 256, 0, stream>>>(hist, nzero);

    score_hist_kernel<<<dim3(NXBLK, NB), 256, 0, stream>>>(logits, obj, hist);
    thresh_kernel<<<NB, 32, 0, stream>>>(hist, thresh);
    compact_kernel<<<dim3(NXBLK, NB), 256, 0, stream>>>(logits, obj, thresh,
                                                        cnt, cscore, cidx);
    select_kernel<<<NB, 1024, 0, stream>>>(boxes, tsz, cnt, cscore, cidx,
                                           (float*)d_out);
}